// TrilinearInterpolation_52501680226537
// MI455X (gfx1250) — compile-verified
//
#include <hip/hip_runtime.h>

// ---------------------------------------------------------------------------
// Trilinear 3D-LUT apply (8 x 3 x 1024 x 1024 image, 3 x 33^3 LUT)
// Memory-bound: ~202 MB traffic -> ~8.7 us floor @ 23.3 TB/s on MI455X.
// Strategy: pack LUT planar->AoS vec4 (8 x b128 gathers/pixel instead of
// 24 scalar), stream x/out with non-temporal TH hints so the 575 KB packed
// LUT stays hot in L2, 128-bit coalesced loads/stores, global_prefetch_b8.
//
// NOTE: the image region of d_out starts at float offset 107811 (after the
// LUT passthrough), i.e. byte offset 431244 == 12 (mod 16). Output stores
// are therefore only 4-byte aligned -> use an align(4) vector type so the
// compiler emits legally-aligned b128 stores (DWORD alignment mode only
// requires 4B alignment for b128 non-formatted ops).
// ---------------------------------------------------------------------------

// Clang vector types (required by __builtin_nontemporal_*; HIP float4 is a
// struct and is rejected).
typedef float v4f   __attribute__((ext_vector_type(4)));              // align 16
typedef float v4f_u __attribute__((ext_vector_type(4), aligned(4)));  // align 4

#define LUT_DIM   33
#define LUT_D2    (LUT_DIM * LUT_DIM)              // 1089
#define LUT_CELLS (LUT_DIM * LUT_DIM * LUT_DIM)    // 35937
#define PLANE     (1024 * 1024)                    // pixels per channel plane
#define NBATCH    8
#define VEC       4                                // pixels per thread
#define BLOCK     256                              // 8 wave32 per block

// --- Pack planar LUT (3, 33,33,33) into AoS v4f[35937] in workspace -------
__global__ __launch_bounds__(BLOCK) void pack_lut_kernel(
    const float* __restrict__ lut, v4f* __restrict__ lutp)
{
    int i = blockIdx.x * BLOCK + threadIdx.x;
    if (i < LUT_CELLS) {
        v4f v;
        v.x = lut[i];
        v.y = lut[LUT_CELLS + i];
        v.z = lut[2 * LUT_CELLS + i];
        v.w = 0.0f;
        lutp[i] = v;
    }
}

// --- Main trilinear kernel ------------------------------------------------
// USE_WS=true : gather packed v4f LUT from d_ws (1 b128 load per corner)
// USE_WS=false: fall back to planar gathers straight from the input LUT
template <bool USE_WS>
__global__ __launch_bounds__(BLOCK) void trilerp_kernel(
    const float* __restrict__ x,      // (8, 3, 1024, 1024), 16B-aligned groups
    const float* __restrict__ lut,    // (3, 33, 33, 33) planar
    const v4f*   __restrict__ lutp,   // packed AoS LUT (may be null)
    float*       __restrict__ out)    // (8, 3, 1024, 1024), only 4B-aligned
{
    const float binsize = 1.000001f / (float)(LUT_DIM - 1);
    const float inv_bin = 1.0f / binsize;

    const int vpp = PLANE / VEC;                       // v4f vectors per plane
    const int v   = blockIdx.x * BLOCK + threadIdx.x;  // global vector id
    const int b   = v / vpp;                           // power-of-2 -> shift
    const int p4  = v - b * vpp;

    const float* xb = x   + (size_t)b * 3 * PLANE + (size_t)p4 * VEC;
    float*       ob = out + (size_t)b * 3 * PLANE + (size_t)p4 * VEC;

    // Prefetch one block-stride ahead in each channel plane (gfx1250
    // global_prefetch_b8; speculative, dropped if past the allocation).
    const int pf = BLOCK * VEC;
    __builtin_prefetch(xb + pf,             0, 1);
    __builtin_prefetch(xb + PLANE + pf,     0, 1);
    __builtin_prefetch(xb + 2 * PLANE + pf, 0, 1);

    // 128-bit non-temporal streaming loads of 4 pixels per channel.
    v4f rv = __builtin_nontemporal_load((const v4f*)(xb));
    v4f gv = __builtin_nontemporal_load((const v4f*)(xb + PLANE));
    v4f bv = __builtin_nontemporal_load((const v4f*)(xb + 2 * PLANE));

    const float rr[VEC] = {rv.x, rv.y, rv.z, rv.w};
    const float gg[VEC] = {gv.x, gv.y, gv.z, gv.w};
    const float bb[VEC] = {bv.x, bv.y, bv.z, bv.w};

    float or_[VEC], og_[VEC], obl_[VEC];

#pragma unroll
    for (int j = 0; j < VEC; ++j) {
        const float fr = rr[j] * inv_bin;
        const float fg = gg[j] * inv_bin;
        const float fb = bb[j] * inv_bin;

        int ri = (int)floorf(fr);
        int gi = (int)floorf(fg);
        int bi = (int)floorf(fb);

        const float rd = fr - (float)ri;
        const float gd = fg - (float)gi;
        const float bd = fb - (float)bi;

        // Clamp for memory safety (x in [0,1) keeps these in range anyway).
        ri = min(max(ri, 0), LUT_DIM - 2);
        gi = min(max(gi, 0), LUT_DIM - 2);
        bi = min(max(bi, 0), LUT_DIM - 2);

        const int id000 = ri + gi * LUT_DIM + bi * LUT_D2;

        const float w000 = (1.0f - rd) * (1.0f - gd) * (1.0f - bd);
        const float w100 = rd          * (1.0f - gd) * (1.0f - bd);
        const float w010 = (1.0f - rd) * gd          * (1.0f - bd);
        const float w110 = rd          * gd          * (1.0f - bd);
        const float w001 = (1.0f - rd) * (1.0f - gd) * bd;
        const float w101 = rd          * (1.0f - gd) * bd;
        const float w011 = (1.0f - rd) * gd          * bd;
        const float w111 = rd          * gd          * bd;

        float accr, accg, accb;

        if (USE_WS) {
            // 8 corner gathers, one global_load_b128 each (all 3 channels).
            const v4f c000 = lutp[id000];
            const v4f c100 = lutp[id000 + 1];
            const v4f c010 = lutp[id000 + LUT_DIM];
            const v4f c110 = lutp[id000 + LUT_DIM + 1];
            const v4f c001 = lutp[id000 + LUT_D2];
            const v4f c101 = lutp[id000 + LUT_D2 + 1];
            const v4f c011 = lutp[id000 + LUT_D2 + LUT_DIM];
            const v4f c111 = lutp[id000 + LUT_D2 + LUT_DIM + 1];

            // Blend all three channels with packed vector FMAs.
            v4f acc = w000 * c000;
            acc += w100 * c100;
            acc += w010 * c010;
            acc += w110 * c110;
            acc += w001 * c001;
            acc += w101 * c101;
            acc += w011 * c011;
            acc += w111 * c111;
            accr = acc.x; accg = acc.y; accb = acc.z;
        } else {
            // Planar fallback: 24 scalar gathers.
            accr = accg = accb = 0.0f;
#pragma unroll
            for (int c = 0; c < 3; ++c) {
                const float* L = lut + (size_t)c * LUT_CELLS;
                const float acc =
                      w000 * L[id000]
                    + w100 * L[id000 + 1]
                    + w010 * L[id000 + LUT_DIM]
                    + w110 * L[id000 + LUT_DIM + 1]
                    + w001 * L[id000 + LUT_D2]
                    + w101 * L[id000 + LUT_D2 + 1]
                    + w011 * L[id000 + LUT_D2 + LUT_DIM]
                    + w111 * L[id000 + LUT_D2 + LUT_DIM + 1];
                if (c == 0) accr = acc; else if (c == 1) accg = acc; else accb = acc;
            }
        }

        or_[j]  = accr;
        og_[j]  = accg;
        obl_[j] = accb;
    }

    // 128-bit non-temporal streaming stores. The output image base is only
    // 4B-aligned (tuple offset 107811 floats), so use the align(4) vector
    // type — legal b128 in DWORD/UNALIGNED alignment modes.
    v4f_u o0 = {or_[0],  or_[1],  or_[2],  or_[3]};
    v4f_u o1 = {og_[0],  og_[1],  og_[2],  og_[3]};
    v4f_u o2 = {obl_[0], obl_[1], obl_[2], obl_[3]};
    __builtin_nontemporal_store(o0, (v4f_u*)(ob));
    __builtin_nontemporal_store(o1, (v4f_u*)(ob + PLANE));
    __builtin_nontemporal_store(o2, (v4f_u*)(ob + 2 * PLANE));
}

extern "C" void kernel_launch(void* const* d_in, const int* in_sizes, int n_in,
                              void* d_out, int out_size, void* d_ws, size_t ws_size,
                              hipStream_t stream)
{
    // Inputs (setup_inputs order): lut_count (int32, unused), lut (f32), x (f32)
    const float* lut = (const float*)d_in[1];
    const float* x   = (const float*)d_in[2];
    float*       out = (float*)d_out;

    // Output tuple = (lut, output): pass the LUT through first.
    (void)hipMemcpyAsync(d_out, (const void*)lut,
                         (size_t)3 * LUT_CELLS * sizeof(float),
                         hipMemcpyDeviceToDevice, stream);

    float* out_img = out + (size_t)3 * LUT_CELLS;

    const int total_vecs = NBATCH * (PLANE / VEC);   // 2,097,152
    const int grid       = total_vecs / BLOCK;       // 8192 (exact)

    const bool use_ws = ws_size >= (size_t)LUT_CELLS * sizeof(v4f);

    if (use_ws) {
        v4f* lutp = (v4f*)d_ws;
        pack_lut_kernel<<<(LUT_CELLS + BLOCK - 1) / BLOCK, BLOCK, 0, stream>>>(lut, lutp);
        trilerp_kernel<true><<<grid, BLOCK, 0, stream>>>(x, lut, lutp, out_img);
    } else {
        trilerp_kernel<false><<<grid, BLOCK, 0, stream>>>(x, lut, nullptr, out_img);
    }
}